// ConfidenceCalibration_15427522527736
// MI455X (gfx1250) — compile-verified
//
#include <hip/hip_runtime.h>
#include <hip/hip_bf16.h>

typedef float f4  __attribute__((ext_vector_type(4)));
typedef float v2f __attribute__((ext_vector_type(2)));
typedef float v8f __attribute__((ext_vector_type(8)));

#define HIST_BLOCKS  2048
#define HIST_THREADS 256
#define QSCALE       16384.0f   // 2^14 fixed-point scale; keeps per-wave partials <= 2^24 (exact in f32/u32)

// Bin b covers (b/nb, (b+1)/nb]; ceil(c*nb)-1. Values outside (0,1] are invalid (dropped overflow bin).
__device__ __forceinline__ int bin_index(float c, int nb, bool& valid) {
    valid = (c > 0.0f) && (c <= 1.0f);
    int idx = (int)ceilf(c * (float)nb) - 1;
    idx = idx < 0 ? 0 : idx;
    idx = idx > nb - 1 ? nb - 1 : idx;
    return idx;
}

__global__ void ece_zero_ws(unsigned long long* __restrict__ ws) {
    ws[threadIdx.x] = 0ull;   // 32 x u64: [0..15]=sum_conf(fixed), [16..31]=sum_acc(fixed)
}

__global__ void ece_hist_kernel(const float* __restrict__ conf,
                                const float* __restrict__ acc,
                                const int*   __restrict__ nbp,
                                unsigned long long* __restrict__ ws,
                                int n) {
    __shared__ unsigned sh[8 * 32];   // 8 per-wave private histograms: [wave][0..15]=conf, [16..31]=acc
    const int tid  = threadIdx.x;
    const int wave = tid >> 5;
    const int lane = tid & 31;

    sh[tid] = 0u;                      // 256 entries, 256 threads
    __syncthreads();

    const int nb = *nbp;
    unsigned* hist = &sh[wave * 32];

    // Streaming phase: float4 loads (global_load_b128), grid-stride.
    const f4* c4 = (const f4*)conf;
    const f4* a4 = (const f4*)acc;
    const int n4     = n >> 2;
    const int stride = (int)(gridDim.x * blockDim.x);
    for (int i = (int)(blockIdx.x * blockDim.x) + tid; i < n4; i += stride) {
        f4 cv = c4[i];
        f4 av = a4[i];
#pragma unroll
        for (int j = 0; j < 4; ++j) {
            float c = cv[j];
            float a = av[j];
            bool valid;
            int  b = bin_index(c, nb, valid);
            if (valid) {
                atomicAdd(&hist[b],      __float2uint_rn(c * QSCALE));  // ds_add_u32
                atomicAdd(&hist[16 + b], __float2uint_rn(a * QSCALE));  // ds_add_u32
            }
        }
    }
    // Tail (n not multiple of 4): handled by wave 0 of block 0.
    const int tail0 = n4 << 2;
    if (blockIdx.x == 0 && tid < (n - tail0)) {
        float c = conf[tail0 + tid];
        float a = acc[tail0 + tid];
        bool valid;
        int  b = bin_index(c, nb, valid);
        if (valid) {
            atomicAdd(&hist[b],      __float2uint_rn(c * QSCALE));
            atomicAdd(&hist[16 + b], __float2uint_rn(a * QSCALE));
        }
    }
    __syncthreads();

    // Block combine: reduce 8 per-wave histograms with V_WMMA_F32_16X16X4_F32.
    // D[m][n] = sum_k 1.0 * B[k][n]; two K=4 steps cover the 8 waves.
    // Wave 0 only (uniform branch -> EXEC all-1s for the WMMA).
    if (tid < 32) {
        const int col   = lane & 15;          // column n = lane%16
        const int khalf = (lane >> 4) << 1;   // lanes 0-15: K=0,1 ; lanes 16-31: K=2,3
        v2f ones = {1.0f, 1.0f};
        v8f dc = {};
        v8f da = {};
#pragma unroll
        for (int s = 0; s < 2; ++s) {
            const int k0 = s * 4 + khalf;
            v2f bc = { (float)sh[(k0 + 0) * 32 + col],
                       (float)sh[(k0 + 1) * 32 + col] };
            v2f ba = { (float)sh[(k0 + 0) * 32 + 16 + col],
                       (float)sh[(k0 + 1) * 32 + 16 + col] };
            dc = __builtin_amdgcn_wmma_f32_16x16x4_f32(false, ones, false, bc,
                                                       (short)0, dc, false, false);
            da = __builtin_amdgcn_wmma_f32_16x16x4_f32(false, ones, false, ba,
                                                       (short)0, da, false, false);
        }
        // D row M=0 lives in VGPR0 of lanes 0-15 (all rows identical since A==1).
        if (lane < 16) {
            atomicAdd(&ws[col],      (unsigned long long)(dc[0] + 0.5f));  // global_atomic_add_u64
            atomicAdd(&ws[16 + col], (unsigned long long)(da[0] + 0.5f));
        }
    }
}

__global__ void ece_finalize_kernel(const unsigned long long* __restrict__ ws,
                                    const int* __restrict__ nbp,
                                    float* __restrict__ out, int n) {
    if (threadIdx.x == 0 && blockIdx.x == 0) {
        int nb = *nbp;
        if (nb > 16) nb = 16;
        float s = 0.0f;
        for (int b = 0; b < nb; ++b) {
            float cs = (float)ws[b]      * (1.0f / QSCALE);
            float as = (float)ws[16 + b] * (1.0f / QSCALE);
            s += fabsf(cs - as);
        }
        out[0] = s / (float)n;
    }
}

extern "C" void kernel_launch(void* const* d_in, const int* in_sizes, int n_in,
                              void* d_out, int out_size, void* d_ws, size_t ws_size,
                              hipStream_t stream) {
    const float* conf = (const float*)d_in[0];
    const float* acc  = (const float*)d_in[1];
    const int*   nbp  = (const int*)d_in[2];
    float* out = (float*)d_out;
    unsigned long long* ws = (unsigned long long*)d_ws;
    const int n = in_sizes[0];

    ece_zero_ws<<<1, 32, 0, stream>>>(ws);

    int n4 = n >> 2;
    int blocks = (n4 + HIST_THREADS - 1) / HIST_THREADS;
    if (blocks > HIST_BLOCKS) blocks = HIST_BLOCKS;
    if (blocks < 1) blocks = 1;
    ece_hist_kernel<<<blocks, HIST_THREADS, 0, stream>>>(conf, acc, nbp, ws, n);

    ece_finalize_kernel<<<1, 1, 0, stream>>>(ws, nbp, out, n);
}